// MultiFrameBoxLoss_7224134991969
// MI455X (gfx1250) — compile-verified
//
#include <hip/hip_runtime.h>
#include <math.h>

// ---------------- problem constants (from reference) ----------------
#define NB   32      // batch
#define NF   8       // frames
#define NA   8732    // anchors
#define NO   32      // truths per frame
#define NEG_POS 3
#define THRESH  0.5f
#define V0      0.1f
#define V1      0.2f
#define NBLK (NB * NF)   // 256 frames, one workgroup each

typedef float v2f __attribute__((ext_vector_type(2)));
typedef float v8f __attribute__((ext_vector_type(8)));

// Wave-level sum reduction through the matrix core:
// A = ones(16x4), B holds the 32 lane values (one per lane in VGPR0 slot,
// zeros elsewhere). D[m,n] = sum_k B[k,n] -> total = sum of the 16 column
// sums = sum of every B entry, regardless of the exact lane->(k,n) striping.
// Finish with 4 xor-shuffle steps. Requires EXEC all-ones.
__device__ __forceinline__ float wave_reduce_wmma(float v) {
  v2f a; a[0] = 1.0f; a[1] = 1.0f;
  v2f b; b[0] = v;    b[1] = 0.0f;
  v8f c = {};
  v8f d = __builtin_amdgcn_wmma_f32_16x16x4_f32(false, a, false, b,
                                                (short)0, c, false, false);
  float s = d[0];                 // every lane holds a column sum of B
  s += __shfl_xor(s, 1, 32);
  s += __shfl_xor(s, 2, 32);
  s += __shfl_xor(s, 4, 32);
  s += __shfl_xor(s, 8, 32);
  return s;
}

// One workgroup per (batch, frame). Dynamic LDS: lc/ov float[NA] + flag int[NA].
__global__ __launch_bounds__(256) void match_frame_kernel(
    const float* __restrict__ loc_data,    // [NB, NF*NA, 4]
    const float* __restrict__ conf_data,   // [NB, NF*NA, 2]
    const float* __restrict__ anchors,     // [NA, 4] center form
    const float* __restrict__ targets,     // [NB, NF, NO, 5]
    float* __restrict__ partials)          // [NBLK, 2]
{
  extern __shared__ char smem_raw[];
  float* s_ovlc = (float*)smem_raw;                        // NA floats
  int*   s_flag = (int*)(smem_raw + (size_t)NA * 4);       // NA ints

  __shared__ float s_tx1[NO], s_ty1[NO], s_tx2[NO], s_ty2[NO], s_lab[NO];
  __shared__ unsigned long long s_bp[NO];   // packed best (iou, ~anchor) per truth
  __shared__ int s_numpos;
  __shared__ float s_red[512];
  __shared__ int s_hist[256];               // radix-select histogram
  __shared__ unsigned s_pref;
  __shared__ int s_krem;

  const int tid = threadIdx.x;
  const int bf  = blockIdx.x;
  const int b   = bf / NF;
  const int f   = bf % NF;

  // ---- load truths for this frame ----
  if (tid < NO * 5) {
    int t = tid / 5, c = tid % 5;
    float v = targets[(((size_t)b * NF + f) * NO + t) * 5 + c];
    if      (c == 0) s_tx1[t] = v;
    else if (c == 1) s_ty1[t] = v;
    else if (c == 2) s_tx2[t] = v;
    else if (c == 3) s_ty2[t] = v;
    else             s_lab[t] = v;
  }
  if (tid < NO) s_bp[tid] = 0ULL;
  if (tid == 0) s_numpos = 0;
  __syncthreads();

  // ---- phase 1: IoU matching ----
  unsigned long long bp[NO];
  #pragma unroll
  for (int t = 0; t < NO; ++t) bp[t] = 0ULL;

  const float4* anc4 = (const float4*)anchors;
  for (int a = tid; a < NA; a += 256) {
    float4 pr = anc4[a];
    float px1 = pr.x - pr.z * 0.5f, py1 = pr.y - pr.w * 0.5f;
    float px2 = pr.x + pr.z * 0.5f, py2 = pr.y + pr.w * 0.5f;
    float ap  = (px2 - px1) * (py2 - py1);
    float best = -1.0f; int bidx = 0;
    #pragma unroll
    for (int t = 0; t < NO; ++t) {
      float tx1 = s_tx1[t], ty1 = s_ty1[t], tx2 = s_tx2[t], ty2 = s_ty2[t];
      float lx = fmaxf(tx1, px1), ly = fmaxf(ty1, py1);
      float rx = fminf(tx2, px2), ry = fminf(ty2, py2);
      float w  = fmaxf(rx - lx, 0.0f), h = fmaxf(ry - ly, 0.0f);
      float inter = w * h;
      float at = (tx2 - tx1) * (ty2 - ty1);
      float iou = inter / (at + ap - inter);
      if (iou > best) { best = iou; bidx = t; }   // first-index wins ties
      unsigned long long key =
          ((unsigned long long)__float_as_uint(iou) << 32) |
          (unsigned long long)(~(unsigned)a);      // lower anchor wins ties
      bp[t] = (key > bp[t]) ? key : bp[t];
    }
    s_ovlc[a] = best;   // best_truth_ov
    s_flag[a] = bidx;   // best_truth_idx
  }
  #pragma unroll
  for (int t = 0; t < NO; ++t) atomicMax(&s_bp[t], bp[t]);
  __syncthreads();

  // ---- forced-match override (serial: last truth wins on duplicates) ----
  if (tid == 0) {
    for (int t = 0; t < NO; ++t) {
      unsigned aidx = ~(unsigned)(s_bp[t] & 0xFFFFFFFFull);
      s_ovlc[aidx] = 2.0f;
      s_flag[aidx] = t;
    }
  }
  __syncthreads();

  // ---- phase 2: encode + loc loss + conf loss (lc) ----
  const float LOGEPS = logf(1e-12f);   // clip floor used by reference BCE
  float pl = 0.0f, pc = 0.0f;
  int my_pos = 0;
  const size_t frame_base = ((size_t)b * NF + f) * NA;
  const float4* loc4  = (const float4*)loc_data;
  const float2* conf2 = (const float2*)conf_data;

  for (int a = tid; a < NA; a += 256) {
    __builtin_prefetch(&conf2[frame_base + a + 256], 0, 0);
    __builtin_prefetch(&loc4 [frame_base + a + 256], 0, 0);

    float ov = s_ovlc[a];
    int   ti = s_flag[a];
    int conf = (ov < THRESH) ? 0 : ((int)s_lab[ti] + 1);
    float2 cc = conf2[frame_base + a];
    float c0 = cc.x, c1 = cc.y;
    int pred = (c1 > c0) ? 1 : 0;          // argmax, first index wins

    if (conf > 0) {
      float4 pr = anc4[a];
      float tx1 = s_tx1[ti], ty1 = s_ty1[ti], tx2 = s_tx2[ti], ty2 = s_ty2[ti];
      float gcx = ((tx1 + tx2) * 0.5f - pr.x) / (V0 * pr.z);
      float gcy = ((ty1 + ty2) * 0.5f - pr.y) / (V0 * pr.w);
      float gw  = logf((tx2 - tx1) / pr.z) / V1;
      float gh  = logf((ty2 - ty1) / pr.w) / V1;
      float4 lv = loc4[frame_base + a];
      float d[4] = {lv.x - gcx, lv.y - gcy, lv.z - gw, lv.w - gh};
      #pragma unroll
      for (int k = 0; k < 4; ++k) {
        float ad = fabsf(d[k]);
        pl += (ad < 1.0f) ? 0.5f * d[k] * d[k] : ad - 0.5f;
      }
      // positive anchors are always selected -> add BCE now
      float tgt    = (float)conf;
      float logp   = (pred == 1) ? 0.0f : LOGEPS;
      float log1mp = (pred == 1) ? LOGEPS : 0.0f;
      pc += -(tgt * logp + (1.0f - tgt) * log1mp);
      my_pos++;
      s_ovlc[a] = 0.0f;   // lc masked to 0 for positives (competes in ranking)
      s_flag[a] = 1;      // positive
    } else {
      float m = fmaxf(c0, c1);
      float lse = m + logf(expf(c0 - m) + expf(c1 - m));
      s_ovlc[a] = lse - c0;             // gathered class 0 for negatives
      s_flag[a] = (pred == 1) ? 2 : 0;  // 2: negative with nonzero BCE
    }
  }
  atomicAdd(&s_numpos, my_pos);
  __syncthreads();

  int num_neg = s_numpos * NEG_POS;
  num_neg = (num_neg < NA - 1) ? num_neg : (NA - 1);

  // ---- phase 3: hard-negative top-k via 4-pass MSB radix select ----
  // All lc >= 0.0f, so unsigned bit-pattern order == numeric order.
  // Find T = num_neg-th largest lc and krem = #slots of value T inside top-k.
  // Stable-argsort equivalence:  selected  <=>  lc > T, or
  //   (lc == T and  #{j < a : lc[j]==T} < krem).
  if (num_neg > 0) {
    if (tid == 0) { s_pref = 0u; s_krem = num_neg; }
    for (int p = 0; p < 4; ++p) {
      __syncthreads();
      s_hist[tid] = 0;                        // blockDim == 256 bins
      __syncthreads();
      unsigned pref = s_pref;
      int shift = 24 - 8 * p;
      for (int a = tid; a < NA; a += 256) {
        unsigned u = __float_as_uint(s_ovlc[a]);
        if (p == 0 || (u >> (shift + 8)) == pref)   // short-circuit avoids >>32
          atomicAdd(&s_hist[(u >> shift) & 255u], 1);
      }
      __syncthreads();
      if (tid == 0) {
        int k = s_krem, cum = 0, bin = 0;
        for (int v = 255; v >= 0; --v) {
          int h = s_hist[v];
          if (cum + h >= k) { bin = v; break; }
          cum += h;
        }
        s_pref = (pref << 8) | (unsigned)bin;
        s_krem = k - cum;                     // >= 1
      }
    }
    __syncthreads();
    const unsigned T = s_pref;
    const int krem = s_krem;

    // Only negatives with pred==1 (flag==2) can contribute to loss_c.
    for (int a = tid; a < NA; a += 256) {
      if (s_flag[a] != 2) continue;
      unsigned u = __float_as_uint(s_ovlc[a]);
      bool sel = false;
      if (u > T) {
        sel = true;
      } else if (u == T) {
        int eq_before = 0;                    // rare path: ties at threshold
        for (int j = 0; j < a; ++j)
          eq_before += (__float_as_uint(s_ovlc[j]) == T) ? 1 : 0;
        sel = (eq_before < krem);
      }
      if (sel) pc += -LOGEPS;   // tgt=0, pred=1 -> bce = -log(1e-12)
    }
  }

  // ---- block reduction (wave0 full -> WMMA reduce) ----
  __syncthreads();
  s_red[tid]       = pl;
  s_red[256 + tid] = pc;
  __syncthreads();
  if (tid < 32) {                          // wave 0, EXEC all-ones
    float vl = 0.0f, vc = 0.0f;
    #pragma unroll
    for (int k = 0; k < 8; ++k) {          // fixed order: deterministic
      vl += s_red[tid + 32 * k];
      vc += s_red[256 + tid + 32 * k];
    }
    float rl = wave_reduce_wmma(vl);
    float rc = wave_reduce_wmma(vc);
    if (tid == 0) {
      partials[2 * bf]     = rl;
      partials[2 * bf + 1] = rc;
    }
  }
}

// Single-wave deterministic final reduction of the 256 per-frame partials.
__global__ __launch_bounds__(32) void finalize_kernel(
    const float* __restrict__ partials, float* __restrict__ out)
{
  int lane = threadIdx.x;
  float vl = 0.0f, vc = 0.0f;
  #pragma unroll
  for (int k = 0; k < 8; ++k) {
    int i = lane + 32 * k;
    vl += partials[2 * i];
    vc += partials[2 * i + 1];
  }
  float rl = wave_reduce_wmma(vl);
  float rc = wave_reduce_wmma(vc);
  if (lane == 0) { out[0] = rl; out[1] = rc; }
}

extern "C" void kernel_launch(void* const* d_in, const int* in_sizes, int n_in,
                              void* d_out, int out_size, void* d_ws, size_t ws_size,
                              hipStream_t stream) {
  const float* loc  = (const float*)d_in[0];   // [32, 8*8732, 4]
  const float* conf = (const float*)d_in[1];   // [32, 8*8732, 2]
  const float* anc  = (const float*)d_in[2];   // [8732, 4]
  const float* tgt  = (const float*)d_in[3];   // [32, 8, 32, 5]
  float* partials = (float*)d_ws;              // 256*2 floats (<< ws_size)
  float* out = (float*)d_out;                  // (loss_l, loss_c)

  size_t shmem = (size_t)NA * (sizeof(float) + sizeof(int));  // ~70 KB of 320 KB
  match_frame_kernel<<<NBLK, 256, shmem, stream>>>(loc, conf, anc, tgt, partials);
  finalize_kernel<<<1, 32, 0, stream>>>(partials, out);
}